// SoftCrossEntropyLossMasking_58497454571825
// MI455X (gfx1250) — compile-verified
//
#include <hip/hip_runtime.h>
#include <stdint.h>

#define AS1 __attribute__((address_space(1)))
#define AS3 __attribute__((address_space(3)))

// Matches the async builtin's parameter pointee type exactly (GCC-style vector):
typedef int v4i_t __attribute__((vector_size(16)));
// Native float4 vector for non-temporal loads / LDS reads:
typedef float f4 __attribute__((ext_vector_type(4)));

constexpr int kB = 16;
constexpr int kS = 4096;
constexpr int kC = 512;
constexpr int WAVES_PER_BLOCK = 8;                       // 256 threads, wave32
constexpr int TOKENS_PER_BLOCK = WAVES_PER_BLOCK;        // one token per wave
constexpr int NCHUNK = kS / TOKENS_PER_BLOCK;            // 512 blocks per batch row
constexpr int NBLOCKS1 = kB * kS / TOKENS_PER_BLOCK;     // 8192 blocks total

// ---- wave32 reductions (gfx1250 is wave32-only) ----
__device__ __forceinline__ float wave_sum(float v) {
#pragma unroll
  for (int i = 16; i > 0; i >>= 1) v += __shfl_xor(v, i, 32);
  return v;
}
__device__ __forceinline__ float wave_max(float v) {
#pragma unroll
  for (int i = 16; i > 0; i >>= 1) v = fmaxf(v, __shfl_xor(v, i, 32));
  return v;
}

__device__ __forceinline__ AS3 v4i_t* lds_ptr(void* p) {
  // low 32 bits of a generic pointer to __shared__ == LDS offset (AS3 ptrs are 32-bit)
  return (AS3 v4i_t*)(uint32_t)(uintptr_t)p;
}
__device__ __forceinline__ AS1 v4i_t* glb_ptr(const void* p) {
  return (AS1 v4i_t*)(uintptr_t)p;
}

// Stage 1: one wave per token. Logits -> registers via non-temporal b128 loads;
// targets -> LDS via async global->LDS (ASYNCcnt path), overlapped with the
// max / exp-sum computation; the wait carries a data dependence on logZ so the
// scheduler cannot hoist it above the softmax-denominator computation.
__global__ __launch_bounds__(256) void ce_stage1(
    const float* __restrict__ logits, const float* __restrict__ target,
    const int* __restrict__ mask, float* __restrict__ loss_part,
    float* __restrict__ cnt_part) {
  __shared__ float lds_t[WAVES_PER_BLOCK][kC];   // 16 KB staged targets
  __shared__ float red_loss[WAVES_PER_BLOCK];
  __shared__ float red_cnt[WAVES_PER_BLOCK];

  const int w = threadIdx.x >> 5;
  const int lane = threadIdx.x & 31;
  const int token = blockIdx.x * TOKENS_PER_BLOCK + w;   // < 65536
  const float* x = logits + (size_t)token * kC;
  const float* t = target + (size_t)token * kC;

  // Kick off async staging of the target row (4 x b128 per lane = 512 floats).
#pragma unroll
  for (int j = 0; j < 4; ++j) {
    const int off = j * 128 + lane * 4;
    __builtin_amdgcn_global_load_async_to_lds_b128(
        glb_ptr(t + off), lds_ptr(&lds_t[w][off]), 0, 0);
  }

  // Logits: 16 floats per lane, streamed once (non-temporal), kept in VGPRs.
  f4 v[4];
#pragma unroll
  for (int j = 0; j < 4; ++j)
    v[j] = __builtin_nontemporal_load((const f4*)(x + j * 128 + lane * 4));

  float m = -INFINITY;
#pragma unroll
  for (int j = 0; j < 4; ++j)
    m = fmaxf(m, fmaxf(fmaxf(v[j].x, v[j].y), fmaxf(v[j].z, v[j].w)));
  m = wave_max(m);

  float e = 0.f;
#pragma unroll
  for (int j = 0; j < 4; ++j)
    e += expf(v[j].x - m) + expf(v[j].y - m) + expf(v[j].z - m) + expf(v[j].w - m);
  e = wave_sum(e);
  const float logZ = m + logf(e);

  // Targets are now needed: wait for this wave's async copies. The "v"(logZ)
  // operand pins this after the logit loads + exp/log work (true overlap).
  asm volatile("s_wait_asynccnt 0" : : "v"(logZ) : "memory");

  float st = 0.f, sx = 0.f;
#pragma unroll
  for (int j = 0; j < 4; ++j) {
    const f4 tv = *(const f4*)(&lds_t[w][j * 128 + lane * 4]);
    st += tv.x + tv.y + tv.z + tv.w;
    sx += tv.x * v[j].x + tv.y * v[j].y + tv.z * v[j].z + tv.w * v[j].w;
  }
  st = wave_sum(st);
  sx = wave_sum(sx);

  if (lane == 0) {
    const int mk = mask[token];
    const float active = (mk == 1) ? 1.f : 0.f;
    red_loss[w] = active * (logZ * st - sx);   // -sum t*logp
    red_cnt[w] = active;
  }
  __syncthreads();
  if (threadIdx.x == 0) {
    float L = 0.f, Cc = 0.f;
#pragma unroll
    for (int i = 0; i < WAVES_PER_BLOCK; ++i) { L += red_loss[i]; Cc += red_cnt[i]; }
    loss_part[blockIdx.x] = L;   // deterministic: each slot written exactly once
    cnt_part[blockIdx.x] = Cc;
  }
}

// Stage 2: per batch row, reduce its 512 block partials.
__global__ __launch_bounds__(256) void ce_stage2(
    const float* __restrict__ loss_part, const float* __restrict__ cnt_part,
    float* __restrict__ batch_sums) {
  const int b = blockIdx.x;
  float L = 0.f, Cc = 0.f;
  for (int i = threadIdx.x; i < NCHUNK; i += 256) {
    L += loss_part[b * NCHUNK + i];
    Cc += cnt_part[b * NCHUNK + i];
  }
  L = wave_sum(L);
  Cc = wave_sum(Cc);
  __shared__ float sL[8], sC[8];
  const int w = threadIdx.x >> 5, lane = threadIdx.x & 31;
  if (lane == 0) { sL[w] = L; sC[w] = Cc; }
  __syncthreads();
  if (threadIdx.x == 0) {
    float a = 0.f, c = 0.f;
#pragma unroll
    for (int i = 0; i < 8; ++i) { a += sL[i]; c += sC[i]; }
    batch_sums[b] = a;
    batch_sums[kB + b] = c;
  }
}

// Stage 3: final scalar on a single wave.
__global__ void ce_stage3(const float* __restrict__ batch_sums,
                          float* __restrict__ out) {
  const int lane = threadIdx.x;
  float num = 0.f, den = 0.f;
  if (lane < kB) {
    const float L = batch_sums[lane];
    const float c = batch_sums[kB + lane];
    const float has = (c > 0.f) ? 1.f : 0.f;
    num = has * (L / fmaxf(c, 1.f));
    den = has;
  }
  num = wave_sum(num);
  den = wave_sum(den);
  if (lane == 0) out[0] = num / fmaxf(den, 1.f);
}

extern "C" void kernel_launch(void* const* d_in, const int* in_sizes, int n_in,
                              void* d_out, int out_size, void* d_ws, size_t ws_size,
                              hipStream_t stream) {
  (void)in_sizes; (void)n_in; (void)out_size; (void)ws_size;
  const float* logits = (const float*)d_in[0];
  const float* target = (const float*)d_in[1];
  const int* mask = (const int*)d_in[2];
  float* out = (float*)d_out;

  float* ws = (float*)d_ws;
  float* loss_part = ws;                    // NBLOCKS1 floats
  float* cnt_part = ws + NBLOCKS1;          // NBLOCKS1 floats
  float* batch_sums = ws + 2 * NBLOCKS1;    // 2*kB floats

  ce_stage1<<<NBLOCKS1, 256, 0, stream>>>(logits, target, mask, loss_part, cnt_part);
  ce_stage2<<<kB, 256, 0, stream>>>(loss_part, cnt_part, batch_sums);
  ce_stage3<<<1, 32, 0, stream>>>(batch_sums, out);
}